// PhysicsConstrainedLoss_38010460570140
// MI455X (gfx1250) — compile-verified
//
#include <hip/hip_runtime.h>

// accum layout in d_ws (doubles):
// 0 bce_sum, 1 mse_sum, 2 S_w, 3 S_wR, 4 S_wX, 5 S_wR2, 6 S_wX2, 7 S_v, 8 S_v2, 9 S_nc2
#define ACC_BYTES 256
#define EPSF 1e-6f

// ---------- CDNA5 async global->LDS copies (DMA-style, tracked by ASYNCcnt) ----------
__device__ __forceinline__ void async_b64(unsigned lds_addr, const void* gaddr) {
    asm volatile("global_load_async_to_lds_b64 %0, %1, off"
                 :: "v"(lds_addr), "v"(gaddr) : "memory");
}
__device__ __forceinline__ void async_b128(unsigned lds_addr, const void* gaddr) {
    asm volatile("global_load_async_to_lds_b128 %0, %1, off"
                 :: "v"(lds_addr), "v"(gaddr) : "memory");
}
template <int N>
__device__ __forceinline__ void wait_asynccnt() {
#if __has_builtin(__builtin_amdgcn_s_wait_asynccnt)
    __builtin_amdgcn_s_wait_asynccnt(N);
#else
    asm volatile("s_wait_asynccnt %0" :: "n"(N) : "memory");
#endif
}

// ---------- per-edge fused math ----------
__device__ __forceinline__ void edge_accum(
    float x, float y, float R, float X, float tR, float tX,
    int s, int d,
    const float* __restrict__ nf, float* __restrict__ node_cur, float* a)
{
    // stable BCE-with-logits: softplus(x) - y*x
    float bce = fmaxf(x, 0.0f) - x * y + log1pf(expf(-fabsf(x)));
    a[0] += bce;

    float dR = R - tR, dX = X - tX;
    a[1] += dR * dR + dX * dX;

    float p = 1.0f / (1.0f + expf(-x));

    const float2* nf2 = (const float2*)nf;       // cols [0,1] of row i at nf2[2*i]
    float2 vs = nf2[2 * (long)s];
    float2 vd = nf2[2 * (long)d];
    float dvr = vs.x - vd.x;
    float dvi = vs.y - vd.y;
    float mag = sqrtf(dvr * dvr + dvi * dvi);
    float rp  = R + EPSF;
    float cur = mag / sqrtf(rp * rp + X * X);
    float curp = cur * p;
    atomicAdd(&node_cur[d],  curp);              // segment_sum(curp, dst)
    atomicAdd(&node_cur[s], -curp);              // - segment_sum(curp, src)

    a[2] += p;
    a[3] += p * R;  a[4] += p * X;
    a[5] += p * R * R;  a[6] += p * X * X;
    float v = mag * p;
    a[7] += v;  a[8] += v * v;
}

// ---------- kernel 0: zero accumulators (every launch: graph-replay safe) ----------
__global__ void zero_kernel(float* __restrict__ node_cur, double* __restrict__ gacc, long N)
{
    long i  = (long)blockIdx.x * blockDim.x + threadIdx.x;
    long gs = (long)gridDim.x * blockDim.x;
    for (long k = i; k < N; k += gs) node_cur[k] = 0.0f;
    if (i < 16) gacc[i] = 0.0;
}

// ---------- kernel 1: streaming edge pass with double-buffered async LDS tiles ----------
// Tile = 512 edges per block-iteration; each of 8 waves owns a 64-edge slice (no barriers).
// Slice layout (2048 B): logits[256] labels[256] src[256] dst[256] params[512] target[512]
__global__ __launch_bounds__(256) void edge_kernel(
    const float* __restrict__ nf,
    const float* __restrict__ logits,
    const float* __restrict__ params,
    const float* __restrict__ labels,
    const float* __restrict__ target,
    const int*   __restrict__ srcA,
    const int*   __restrict__ dstA,
    long E, float* __restrict__ node_cur, double* __restrict__ gacc)
{
    __shared__ char smem[8 * 2 * 2048];          // 32 KB of the 320 KB WGP LDS

    float a[9];
#pragma unroll
    for (int i = 0; i < 9; ++i) a[i] = 0.0f;

    const int lane = threadIdx.x & 31;
    const int wave = threadIdx.x >> 5;
    char* wbase = smem + wave * 4096;
    // low 32 bits of a generic LDS address == workgroup-relative LDS byte offset
    const unsigned wbase_lds = (unsigned)(unsigned long long)(void*)wbase;
    const unsigned o8  = (unsigned)lane * 8u;
    const unsigned o16 = (unsigned)lane * 16u;

    const long ntiles = E >> 9;                  // 512-edge tiles

    auto issue = [&](int b, long t) {
        long e0 = (t << 9) + (long)wave * 64;    // first edge of this wave's slice
        unsigned lb = wbase_lds + (unsigned)(b * 2048);
        async_b64 (lb + 0    + o8,  (const char*)(logits + e0) + o8);
        async_b64 (lb + 256  + o8,  (const char*)(labels + e0) + o8);
        async_b64 (lb + 512  + o8,  (const char*)(srcA   + e0) + o8);
        async_b64 (lb + 768  + o8,  (const char*)(dstA   + e0) + o8);
        async_b128(lb + 1024 + o16, (const char*)(params + 2 * e0) + o16);
        async_b128(lb + 1536 + o16, (const char*)(target + 2 * e0) + o16);
    };

    long tile = blockIdx.x;
    int  buf  = 0;
    if (tile < ntiles) issue(buf, tile);
    while (tile < ntiles) {
        long nxt = tile + (long)gridDim.x;
        if (nxt < ntiles) { issue(buf ^ 1, nxt); wait_asynccnt<6>(); }
        else              { wait_asynccnt<0>(); }

        char* pb = wbase + buf * 2048;
        float2 xl = *(const float2*)(pb + 0    + (size_t)lane * 8);
        float2 yl = *(const float2*)(pb + 256  + (size_t)lane * 8);
        int2   sl = *(const int2*)  (pb + 512  + (size_t)lane * 8);
        int2   dl = *(const int2*)  (pb + 768  + (size_t)lane * 8);
        float4 pp = *(const float4*)(pb + 1024 + (size_t)lane * 16);
        float4 tt = *(const float4*)(pb + 1536 + (size_t)lane * 16);

        edge_accum(xl.x, yl.x, pp.x, pp.y, tt.x, tt.y, sl.x, dl.x, nf, node_cur, a);
        edge_accum(xl.y, yl.y, pp.z, pp.w, tt.z, tt.w, sl.y, dl.y, nf, node_cur, a);

        tile = nxt;
        buf ^= 1;
    }

    // tail (E not a multiple of 512): plain global loads
    {
        long gid = (long)blockIdx.x * blockDim.x + threadIdx.x;
        long gsz = (long)gridDim.x * blockDim.x;
        for (long e = (ntiles << 9) + gid; e < E; e += gsz) {
            edge_accum(logits[e], labels[e],
                       params[2 * e], params[2 * e + 1],
                       target[2 * e], target[2 * e + 1],
                       srcA[e], dstA[e], nf, node_cur, a);
        }
    }

    // wave32 butterfly reduction, then one f64 atomic per wave per quantity
#pragma unroll
    for (int m = 16; m >= 1; m >>= 1) {
#pragma unroll
        for (int i = 0; i < 9; ++i) a[i] += __shfl_xor(a[i], m, 32);
    }
    if (lane == 0) {
#pragma unroll
        for (int i = 0; i < 9; ++i) atomicAdd(&gacc[i], (double)a[i]);
    }
}

// ---------- kernel 2: sum of node_cur^2 ----------
__global__ __launch_bounds__(256) void node_kernel(
    const float* __restrict__ node_cur, long N, double* __restrict__ gacc)
{
    float s = 0.0f;
    long gid = (long)blockIdx.x * blockDim.x + threadIdx.x;
    long gsz = (long)gridDim.x * blockDim.x;
    for (long i = gid; i < N; i += gsz) {
        float v = node_cur[i];
        s += v * v;
    }
#pragma unroll
    for (int m = 16; m >= 1; m >>= 1) s += __shfl_xor(s, m, 32);
    if ((threadIdx.x & 31) == 0) atomicAdd(&gacc[9], (double)s);
}

// ---------- kernel 3: combine scalars ----------
__global__ void finalize_kernel(const double* __restrict__ g, float* __restrict__ out,
                                long E, long N)
{
    if (threadIdx.x == 0 && blockIdx.x == 0) {
        const double dE = (double)E;
        double topology  = 1.0  * (g[0] / dE);
        double parameter = 1.0  * (g[1] / (2.0 * dE));
        double kcl       = 0.1  * (g[9] / (double)N);
        double Sw = g[2];
        double mR = g[3] / (Sw + 1e-6);
        double mX = g[4] / (Sw + 1e-6);
        double varR = g[5] - 2.0 * mR * g[3] + mR * mR * Sw;
        double varX = g[6] - 2.0 * mX * g[4] + mX * mX * Sw;
        double param_consistency = 0.5 * (varR + varX);
        double volt_consistency  = (g[8] - g[7] * g[7] / dE) / (dE - 1.0);
        double kvl = 0.1 * (param_consistency + volt_consistency);
        double ec  = Sw - (double)(N - 1);
        double pmean = Sw / dE;
        double radial   = 0.1 * (ec * ec + 0.1 * pmean);
        double sparsity = 0.01 * pmean;
        out[0] = (float)topology;
        out[1] = (float)parameter;
        out[2] = (float)kcl;
        out[3] = (float)kvl;
        out[4] = (float)radial;
        out[5] = (float)sparsity;
        out[6] = (float)(topology + parameter + kcl + kvl + radial + sparsity);
    }
}

extern "C" void kernel_launch(void* const* d_in, const int* in_sizes, int n_in,
                              void* d_out, int out_size, void* d_ws, size_t ws_size,
                              hipStream_t stream)
{
    const float* nf     = (const float*)d_in[0];   // (N,4) f32
    const float* logits = (const float*)d_in[1];   // (E,)  f32
    const float* params = (const float*)d_in[2];   // (E,2) f32
    const float* labels = (const float*)d_in[3];   // (E,)  f32
    const float* target = (const float*)d_in[4];   // (E,2) f32
    const int*   eidx   = (const int*)d_in[5];     // (2,E) i32

    const long E = (long)in_sizes[1];
    const long N = (long)in_sizes[0] / 4;
    const int* srcA = eidx;
    const int* dstA = eidx + E;

    double* gacc     = (double*)d_ws;
    float*  node_cur = (float*)((char*)d_ws + ACC_BYTES);
    float*  out      = (float*)d_out;

    int zblocks = (int)((N + 255) / 256);
    zero_kernel<<<zblocks, 256, 0, stream>>>(node_cur, gacc, N);
    edge_kernel<<<1024, 256, 0, stream>>>(nf, logits, params, labels, target,
                                          srcA, dstA, E, node_cur, gacc);
    node_kernel<<<512, 256, 0, stream>>>(node_cur, N, gacc);
    finalize_kernel<<<1, 32, 0, stream>>>(gacc, out, E, N);
}